// Weighted_SISTA_11931419148441
// MI455X (gfx1250) — compile-verified
//
#include <hip/hip_runtime.h>
#include <hip/hip_bf16.h>

// Weighted SISTA for MI455X (gfx1250, wave32, WMMA f32 16x16x4 + TDM).
// BATCH=256, DIM=256, M=64, T=256, NUM_ITERS=4.

typedef float v2f __attribute__((ext_vector_type(2)));
typedef float v8f __attribute__((ext_vector_type(8)));
typedef unsigned int u32x4 __attribute__((ext_vector_type(4)));
typedef int i32x4 __attribute__((ext_vector_type(4)));
typedef int i32x8 __attribute__((ext_vector_type(8)));

#define DIMN  256
#define MDIM  64
#define TLEN  256
#define BSZ   256
#define NITER 4
#define HSTR  266   // LDS row stride for h buffers (mod 64 == 10: conflict-safe)
#define XSTR  74    // LDS row stride for x staging (TDM pad: 64 + 10 DWORDs)

__device__ __forceinline__ v8f wmma4(v2f a, v2f b, v8f c) {
  // D = A(16x4 f32) x B(4x16 f32) + C(16x16 f32)
  return __builtin_amdgcn_wmma_f32_16x16x4_f32(false, a, false, b, (short)0, c,
                                               false, false);
}

// Tensor Data Mover: async-load a 16x64 f32 tile (row stride 64 elems in
// memory) into LDS at lds_off, padding each 64-DWORD row with 10 DWORDs so
// the LDS image has row stride XSTR=74 (bank-conflict-safe for WMMA A loads).
__device__ __forceinline__ void tdm_load_x_tile(unsigned lds_off,
                                                const float* gptr) {
  unsigned long long ga = (unsigned long long)(uintptr_t)gptr;
  u32x4 g0;
  g0[0] = 1u;                                   // count=1 (valid user D#)
  g0[1] = lds_off;                              // LDS byte address
  g0[2] = (unsigned)(ga & 0xFFFFFFFFu);         // global_addr[31:0]
  g0[3] = (unsigned)((ga >> 32) & 0x1FFFFFFu)   // global_addr[56:32]
          | (2u << 30);                         // type = 2 ("image")
  i32x8 g1;
  // data_size=2 (4B) | pad_enable | pad_interval=5 (64 DW) | pad_amount=9 (10 DW)
  g1[0] = (int)((2u << 16) | (1u << 20) | (5u << 22) | (9u << 25));
  g1[1] = (int)(64u << 16);                     // tensor_dim0 = 64 (bits 79:48)
  g1[2] = 0;                                    // dim0 hi=0, tensor_dim1 lo=0
  g1[3] = (int)(1u | (64u << 16));              // tensor_dim1=65536 hi, tile_dim0=64
  g1[4] = 16;                                   // tile_dim1 = 16 rows
  g1[5] = 64;                                   // tensor_dim0_stride = 64 elems
  g1[6] = 0;
  g1[7] = 0;
  i32x4 g2 = {0, 0, 0, 0};                      // 2-D tile: groups 2/3 unused
  i32x4 g3 = {0, 0, 0, 0};
#if defined(__clang_major__) && __clang_major__ >= 23
  i32x8 gz = {0, 0, 0, 0, 0, 0, 0, 0};
  __builtin_amdgcn_tensor_load_to_lds(g0, g1, g2, g3, gz, 0);
#else
  __builtin_amdgcn_tensor_load_to_lds(g0, g1, g2, g3, 0);
#endif
}

// ---------------- precompute (tiny, plain FMA) ----------------
// FD = F @ D ; T1 = A^T A + lam2 I ; Vb[m][d] = (1/alpha) * sum_k D[k][d] A[m][k]
__global__ __launch_bounds__(256)
void sista_pre1(const float* __restrict__ Ag, const float* __restrict__ Dg,
                const float* __restrict__ Fg, const float* __restrict__ lam2p,
                const float* __restrict__ alphap,
                float* __restrict__ FD, float* __restrict__ T1,
                float* __restrict__ Vb) {
  int i = blockIdx.x * blockDim.x + threadIdx.x;   // 0 .. 147455
  if (i < 65536) {
    int r = i >> 8, c = i & 255;
    float s = 0.f;
    for (int k = 0; k < DIMN; ++k) s = fmaf(Fg[r*DIMN+k], Dg[k*DIMN+c], s);
    FD[i] = s;
  } else if (i < 131072) {
    int j = i - 65536; int r = j >> 8, c = j & 255;
    float s = (r == c) ? lam2p[0] : 0.f;
    for (int m = 0; m < MDIM; ++m) s = fmaf(Ag[m*DIMN+r], Ag[m*DIMN+c], s);
    T1[j] = s;
  } else {
    int j = i - 131072; int m = j >> 8, d = j & 255;
    float s = 0.f;
    for (int k = 0; k < DIMN; ++k) s = fmaf(Dg[k*DIMN+d], Ag[m*DIMN+k], s);
    Vb[j] = s * (1.0f / alphap[0]);
  }
}

// Pb[k][n] = P[n][k] = sum_j D[j][n]*FD[j][k] ; T1D = T1 @ D
__global__ __launch_bounds__(256)
void sista_pre2(const float* __restrict__ Dg, const float* __restrict__ FD,
                const float* __restrict__ T1,
                float* __restrict__ Pb, float* __restrict__ T1D) {
  int i = blockIdx.x * blockDim.x + threadIdx.x;   // 0 .. 131071
  if (i < 65536) {
    int k = i >> 8, n = i & 255;
    float s = 0.f;
    for (int j = 0; j < DIMN; ++j) s = fmaf(Dg[j*DIMN+n], FD[j*DIMN+k], s);
    Pb[i] = s;
  } else {
    int j2 = i - 65536; int r = j2 >> 8, c = j2 & 255;
    float s = 0.f;
    for (int k = 0; k < DIMN; ++k) s = fmaf(T1[r*DIMN+k], Dg[k*DIMN+c], s);
    T1D[j2] = s;
  }
}

// Sb[k][n] = S[n][k] = delta(n,k) - (1/alpha) * sum_j D[j][n]*T1D[j][k]
__global__ __launch_bounds__(256)
void sista_pre3(const float* __restrict__ Dg, const float* __restrict__ T1D,
                const float* __restrict__ alphap, float* __restrict__ Sb) {
  int i = blockIdx.x * blockDim.x + threadIdx.x;   // 0 .. 65535
  int k = i >> 8, n = i & 255;
  float s = 0.f;
  for (int j = 0; j < DIMN; ++j) s = fmaf(Dg[j*DIMN+n], T1D[j*DIMN+k], s);
  Sb[i] = ((n == k) ? 1.0f : 0.0f) - s * (1.0f / alphap[0]);
}

// ---------------- x projection: x[t][b][m] = sum_d A[m][d] * y[b][d][t] ----
// WMMA orientation: M=t-tile (coalesced y reads), N=m-tile, K=d.
__global__ __launch_bounds__(256)
void sista_xproj(const float* __restrict__ Ag, const float* __restrict__ yg,
                 float* __restrict__ xg) {
  const int wid  = threadIdx.x >> 5;
  const int lane = threadIdx.x & 31;
  const int hi = lane >> 4, lr = lane & 15;
  const int tile = blockIdx.x * 8 + wid;          // 16384 tiles
  const int b  = tile >> 6;
  const int tt = (tile >> 2) & 15;
  const int mt = tile & 3;
  const int t0 = tt * 16, m0 = mt * 16;

  v8f acc = {};
  #pragma unroll 8
  for (int kk = 0; kk < DIMN; kk += 4) {
    const int ka = kk + 2 * hi;
    v2f a;
    a.x = yg[((size_t)b * DIMN + ka    ) * TLEN + t0 + lr];
    a.y = yg[((size_t)b * DIMN + ka + 1) * TLEN + t0 + lr];
    v2f bb = *(const v2f*)(Ag + (m0 + lr) * DIMN + ka);
    acc = wmma4(a, bb, acc);
  }
  #pragma unroll
  for (int j = 0; j < 8; ++j)
    xg[((size_t)(t0 + j + 8 * hi) * BSZ + b) * MDIM + m0 + lr] = acc[j];
}

// ---------------- sequential scan ----------------
// 16 workgroups x 512 threads (16 waves). Each WG owns 16 batch rows for all
// T steps (no global sync). Wave w computes dim-columns [16w, 16w+16).
// S^T / V^T B-operand slices are kernel-invariant -> VGPR-resident (128+32).
// x tiles arrive via the Tensor Data Mover, double-buffered one step ahead
// (TENSORcnt + barrier), overlapping the global->LDS copy with the inner ISTA GEMMs.
__global__ __launch_bounds__(512)
void sista_scan(const float* __restrict__ xg,    // [T][B][M]
                const float* __restrict__ Pb,    // (k,n) = P[n][k]
                const float* __restrict__ Sb,    // (k,n) = S[n][k]
                const float* __restrict__ Vb,    // (m,n) = V[n][m]
                const float* __restrict__ h0,
                const float* __restrict__ Ug,
                const float* __restrict__ lam1p,
                const float* __restrict__ lam2p,
                const float* __restrict__ alphap,
                float* __restrict__ hs)          // [B][T][DIM]
{
  __shared__ float hb[3][16 * HSTR];
  __shared__ float xs[2][16 * XSTR];

  const int tid  = threadIdx.x;
  const int wid  = tid >> 5;
  const int lane = tid & 31;
  const int hi = lane >> 4, lr = lane & 15;
  const int n0 = wid * 16;
  const int b0 = blockIdx.x * 16;

  const float inva = 1.0f / alphap[0];
  const float bthr = lam1p[0] * inva;
  const float c2   = lam2p[0] * inva;
  const float ub   = Ug[n0 + lr] * bthr;   // threshold for this wave's column

  // ---- hoist the wave-invariant B-operand slices into registers ----
  v2f sB[64];                               // S^T slice: K=256 -> 128 VGPRs
  #pragma unroll
  for (int k4 = 0; k4 < 64; ++k4) {
    const int ka = 4 * k4 + 2 * hi;
    sB[k4].x = Sb[(ka    ) * DIMN + n0 + lr];
    sB[k4].y = Sb[(ka + 1) * DIMN + n0 + lr];
  }
  v2f vB[16];                               // V^T slice: K=64 -> 32 VGPRs
  #pragma unroll
  for (int k4 = 0; k4 < 16; ++k4) {
    const int ka = 4 * k4 + 2 * hi;
    vB[k4].x = Vb[(ka    ) * DIMN + n0 + lr];
    vB[k4].y = Vb[(ka + 1) * DIMN + n0 + lr];
  }

  // stage h0 into buffer 0 (h_pre for t=0); kick off TDM for x tile t=0
  for (int i = tid; i < 16 * 256; i += 512) {
    int r = i >> 8, c = i & 255;
    hb[0][r * HSTR + c] = h0[c];
  }
  if (wid == 0)
    tdm_load_x_tile((unsigned)(uintptr_t)&xs[0][0], xg + (size_t)b0 * MDIM);

  int p = 0, q = 1, r2 = 2;

  for (int t = 0; t < TLEN; ++t) {
    // x tile for this step was TDM'd one step ahead; publish it.
    if (wid == 0) __builtin_amdgcn_s_wait_tensorcnt((short)0);
    __syncthreads();

    // warm L2 for the x tile after next (global_prefetch_b8)
    if (t + 2 < TLEN)
      __builtin_prefetch(xg + ((size_t)(t + 2) * BSZ + b0) * MDIM + tid * 2, 0, 0);

    // phase 1: Ph = h_pre @ P^T   (K = 256, P streamed from L2)
    v8f ph = {};
    {
      const float* hsrc = &hb[p][lr * HSTR];
      #pragma unroll 8
      for (int kk = 0; kk < DIMN; kk += 4) {
        const int ka = kk + 2 * hi;
        v2f a = *(const v2f*)(hsrc + ka);
        v2f b;
        b.x = Pb[(ka    ) * DIMN + n0 + lr];
        b.y = Pb[(ka + 1) * DIMN + n0 + lr];
        ph = wmma4(a, b, ph);
      }
    }
    // xv = x_t @ V^T   (K = 64, B from registers)
    v8f xv = {};
    {
      const float* xsrc = &xs[t & 1][lr * XSTR];
      #pragma unroll
      for (int k4 = 0; k4 < 16; ++k4) {
        v2f a = *(const v2f*)(xsrc + 4 * k4 + 2 * hi);
        xv = wmma4(a, vB[k4], xv);
      }
    }
    // x tile consumed -> issue TDM for step t+1 into the other buffer;
    // it overlaps the 4 inner iterations below.
    if (wid == 0 && t + 1 < TLEN)
      tdm_load_x_tile((unsigned)(uintptr_t)&xs[(t + 1) & 1][0],
                      xg + ((size_t)(t + 1) * BSZ + b0) * MDIM);

    float cst[8];
    #pragma unroll
    for (int j = 0; j < 8; ++j) cst[j] = xv[j] + c2 * ph[j];

    // h_iter(0) = Ph  -> work buffer q
    #pragma unroll
    for (int j = 0; j < 8; ++j)
      hb[q][(j + 8 * hi) * HSTR + n0 + lr] = ph[j];
    __syncthreads();

    // 4 inner ISTA iterations, double-buffered q<->r2.
    // Inner GEMM: 64 x (ds_load_b64 + v_wmma), B entirely in VGPRs.
    int src = q, dst = r2;
    #pragma unroll
    for (int it = 0; it < NITER; ++it) {
      v8f acc = {};
      const float* hsrc = &hb[src][lr * HSTR];
      #pragma unroll
      for (int k4 = 0; k4 < 64; ++k4) {
        v2f a = *(const v2f*)(hsrc + 4 * k4 + 2 * hi);
        acc = wmma4(a, sB[k4], acc);
      }
      float* d = hb[dst];
      #pragma unroll
      for (int j = 0; j < 8; ++j) {
        float z = acc[j] + cst[j];
        float m = __builtin_fabsf(z) - ub;
        float rr = (m > 0.f) ? ((z < 0.f) ? -m : m) : 0.f;
        d[(j + 8 * hi) * HSTR + n0 + lr] = rr;
        if (it == NITER - 1)   // final iterate -> hs[b][t][dim], 64B coalesced
          hs[((size_t)(b0 + j + 8 * hi) * TLEN + t) * DIMN + n0 + lr] = rr;
      }
      if (it != NITER - 1) __syncthreads();   // last one folded into step-top barrier
      int tmp = src; src = dst; dst = tmp;
    }
    // h_t landed in buffer q; it becomes h_pre, old h_pre becomes scratch
    int tmp = p; p = q; q = tmp;
  }
}

// ---------------- reconstruction: out[b][n][t] = sum_d D[n][d] hs[b][t][d] --
__global__ __launch_bounds__(256)
void sista_recon(const float* __restrict__ Dg, const float* __restrict__ hs,
                 float* __restrict__ outp) {
  const int wid  = threadIdx.x >> 5;
  const int lane = threadIdx.x & 31;
  const int hi = lane >> 4, lr = lane & 15;
  const int tile = blockIdx.x * 8 + wid;          // 65536 tiles
  const int b  = tile >> 8;
  const int nt = (tile >> 4) & 15;
  const int tt = tile & 15;
  const int n0 = nt * 16, t0 = tt * 16;

  v8f acc = {};
  #pragma unroll 8
  for (int kk = 0; kk < DIMN; kk += 4) {
    const int ka = kk + 2 * hi;
    v2f a  = *(const v2f*)(Dg + (n0 + lr) * DIMN + ka);
    v2f bb = *(const v2f*)(hs + ((size_t)b * TLEN + t0 + lr) * DIMN + ka);
    acc = wmma4(a, bb, acc);
  }
  #pragma unroll
  for (int j = 0; j < 8; ++j)
    outp[((size_t)b * DIMN + n0 + j + 8 * hi) * TLEN + t0 + lr] = acc[j];
}

extern "C" void kernel_launch(void* const* d_in, const int* in_sizes, int n_in,
                              void* d_out, int out_size, void* d_ws, size_t ws_size,
                              hipStream_t stream) {
  const float* y     = (const float*)d_in[0];   // (B, DIM, T)
  const float* A     = (const float*)d_in[1];   // (M, DIM)
  const float* D     = (const float*)d_in[2];   // (DIM, DIM)
  const float* F     = (const float*)d_in[3];   // (DIM, DIM)
  const float* lam1  = (const float*)d_in[4];
  const float* lam2  = (const float*)d_in[5];
  const float* alpha = (const float*)d_in[6];
  const float* h0    = (const float*)d_in[7];   // (DIM,)
  const float* U     = (const float*)d_in[8];   // (DIM,)
  float* out = (float*)d_out;
  float* ws  = (float*)d_ws;

  float* Sb  = ws;                 // 65536
  float* Pb  = ws + 65536;         // 65536
  float* Vb  = ws + 131072;        // 16384
  float* FD  = ws + 147456;        // 65536
  float* T1  = ws + 212992;        // 65536
  float* T1D = ws + 278528;        // 65536
  float* xb  = ws + 344064;        // T*B*M = 4194304
  float* hsb = ws + 4538368;       // B*T*DIM = 16777216   (total ~85 MB)

  sista_pre1<<<576, 256, 0, stream>>>(A, D, F, lam2, alpha, FD, T1, Vb);
  sista_pre2<<<512, 256, 0, stream>>>(D, FD, T1, Pb, T1D);
  sista_pre3<<<256, 256, 0, stream>>>(D, T1D, alpha, Sb);
  sista_xproj<<<2048, 256, 0, stream>>>(A, y, xb);
  sista_scan<<<16, 512, 0, stream>>>(xb, Pb, Sb, Vb, h0, U, lam1, lam2, alpha, hsb);
  sista_recon<<<8192, 256, 0, stream>>>(D, hsb, out);
}